// RnRoPE_20298015441194
// MI455X (gfx1250) — compile-verified
//
#include <hip/hip_runtime.h>
#include <hip/hip_bf16.h>

typedef __attribute__((ext_vector_type(2))) float v2f;
typedef __attribute__((ext_vector_type(4))) float f4v;
typedef __attribute__((ext_vector_type(8))) float v8f;
typedef __attribute__((ext_vector_type(4))) int   i4v;

#define BB 4
#define HH 16
#define SS 4096
#define DD 64
#define PAD 68   // LDS row stride in floats (4-float pad -> conflict-free strided access)

#if defined(__has_builtin)
#  if __has_builtin(__builtin_amdgcn_global_load_async_to_lds_b128)
#    define ASYNC_LDS 1
#  endif
#  if __has_builtin(__builtin_amdgcn_s_wait_asynccnt)
#    define WAIT_ASYNC() __builtin_amdgcn_s_wait_asynccnt(0)
#  endif
#endif
#ifndef WAIT_ASYNC
#  define WAIT_ASYNC() asm volatile("s_wait_asynccnt 0x0" ::: "memory")
#endif

// b128 async copy moves one int4 per lane: pointers are int4* in AS1 (global) / AS3 (LDS)
typedef __attribute__((address_space(1))) i4v* gas_ptr;
typedef __attribute__((address_space(3))) i4v* lds_ptr;

// ---------------------------------------------------------------------------
// Kernel 1: build 64x64 Householder product Q in d_ws (row-major).
// One block, 64 threads; thread j owns column j of Q (updates are column-private).
// ---------------------------------------------------------------------------
__global__ __launch_bounds__(64) void RnRoPE_build_Q(const float* __restrict__ vs,
                                                     float* __restrict__ Qg) {
    __shared__ float Qc[64][64];   // Qc[i][j]
    __shared__ float vsh[64];
    const int j = threadIdx.x;
    #pragma unroll 4
    for (int i = 0; i < 64; ++i) Qc[i][j] = (i == j) ? 1.0f : 0.0f;
    __syncthreads();
    for (int r = 0; r < 32; ++r) {
        vsh[j] = vs[r * 64 + j];
        __syncthreads();
        float w = 0.0f, nv = 0.0f;
        #pragma unroll 4
        for (int i = 0; i < 64; ++i) {
            float vi = vsh[i];
            w  += vi * Qc[i][j];
            nv += vi * vi;
        }
        float coef = 2.0f / (nv + 1e-8f);
        #pragma unroll 4
        for (int i = 0; i < 64; ++i) Qc[i][j] -= coef * vsh[i] * w;
        __syncthreads();
    }
    #pragma unroll 4
    for (int i = 0; i < 64; ++i) Qg[i * 64 + j] = Qc[i][j];
}

// ---------------------------------------------------------------------------
// Kernel 2: out = ((x @ Q^T) * cos + rotate_half(x @ Q^T) * sin) @ Q
// for x in {q, k}. 16-row tiles per wave32, fp32 WMMA 16x16x4,
// async global->LDS staging (ASYNCcnt path).
// ---------------------------------------------------------------------------
__global__ __launch_bounds__(256) void RnRoPE_main(const float* __restrict__ q,
                                                   const float* __restrict__ k,
                                                   const float* __restrict__ cosp,
                                                   const float* __restrict__ sinp,
                                                   const float* __restrict__ Qg,
                                                   float* __restrict__ out) {
    __shared__ float Qs[64 * PAD];        // Q   row-major  (B-frags for matmul1: B = Q^T)
    __shared__ float Qt[64 * PAD];        // Q^T row-major  (B-frags for matmul2: B = Q)
    __shared__ float Ts[8][16 * PAD];     // per-wave staging tile (16 x 64, padded)

    const int tid = threadIdx.x;
    // Cooperative load of Q into Qs and its transpose into Qt.
    for (int i = tid; i < 64 * 64; i += 256) {
        int r = i >> 6, c = i & 63;
        float v = Qg[i];
        Qs[r * PAD + c] = v;
        Qt[c * PAD + r] = v;
    }
    __syncthreads();

    const int wave = tid >> 5;
    const int lane = tid & 31;
    const int lo   = lane & 15;   // lane within half-wave
    const int hi   = lane >> 4;   // which half-wave
    float* ts = Ts[wave];

    const long tensorElems = (long)BB * HH * SS * DD;   // 16,777,216

    for (int it = 0; it < 4; ++it) {
        const int tglob  = (blockIdx.x * 8 + wave) * 4 + it;   // 0..32767
        const int tensor = tglob >> 14;                        // 0 = q, 1 = k
        const int tile   = tglob & 16383;
        const int r0     = tile << 4;                          // first row in tensor
        const int b      = r0 >> 16;                           // / (H*S)
        const int s0     = r0 & (SS - 1);                      // seq position of row 0

        const float* gsrc  = (tensor ? k : q) + (long)r0 * DD;
        float*       gdst  = out + (long)tensor * tensorElems + (long)r0 * DD;
        const float* cbase = cosp + ((long)b * SS + s0) * DD;
        const float* sbase = sinp + ((long)b * SS + s0) * DD;

        // ---- stage 16x64 input tile into wave-private LDS ----
        __builtin_prefetch(gsrc + 16 * DD, 0, 0);   // speculative prefetch of next tile
#ifdef ASYNC_LDS
        // WAR: previous iteration's ds_loads of ts use DScnt; async writes use
        // ASYNCcnt with no cross-counter ordering -> drain DS first.
        asm volatile("s_wait_dscnt 0x0" ::: "memory");
        #pragma unroll
        for (int i = 0; i < 8; ++i) {
            int off = i * 128 + lane * 4;           // flat float offset in tile
            int r = off >> 6, c = off & 63;
            __builtin_amdgcn_global_load_async_to_lds_b128(
                (gas_ptr)(gsrc + off), (lds_ptr)(ts + r * PAD + c), 0, 0);
        }
        WAIT_ASYNC();                               // LDS contents valid for ds_load
#else
        f4v tmp[8];
        #pragma unroll
        for (int i = 0; i < 8; ++i)
            tmp[i] = *(const f4v*)(gsrc + i * 128 + lane * 4);
        #pragma unroll
        for (int i = 0; i < 8; ++i) {
            int off = i * 128 + lane * 4;
            int r = off >> 6, c = off & 63;
            *(f4v*)(ts + r * PAD + c) = tmp[i];
        }
#endif

        // ---- matmul1: acc[n] = tile @ Q^T  (B[k][n] = Q[16n+lo][k]) ----
        v8f a0 = {}, a1 = {}, a2 = {}, a3 = {};
        #pragma unroll
        for (int kk = 0; kk < 16; ++kk) {
            const int colA = 4 * kk + 2 * hi;       // even -> 8B-aligned float2
            v2f af = *(const v2f*)(ts + lo * PAD + colA);
            v2f b0 = *(const v2f*)(Qs + (0 * 16 + lo) * PAD + colA);
            v2f b1 = *(const v2f*)(Qs + (1 * 16 + lo) * PAD + colA);
            v2f b2 = *(const v2f*)(Qs + (2 * 16 + lo) * PAD + colA);
            v2f b3 = *(const v2f*)(Qs + (3 * 16 + lo) * PAD + colA);
            a0 = __builtin_amdgcn_wmma_f32_16x16x4_f32(false, af, false, b0, (short)0, a0, false, false);
            a1 = __builtin_amdgcn_wmma_f32_16x16x4_f32(false, af, false, b1, (short)0, a1, false, false);
            a2 = __builtin_amdgcn_wmma_f32_16x16x4_f32(false, af, false, b2, (short)0, a2, false, false);
            a3 = __builtin_amdgcn_wmma_f32_16x16x4_f32(false, af, false, b3, (short)0, a3, false, false);
        }

        // ---- RoPE in WMMA C/D layout; rotate_half is acc-tile swap n <-> n^2 ----
        // element (vgpr v, lane) = qi[row = v + 8*hi][d = 16n + lo]
        #pragma unroll
        for (int v = 0; v < 8; ++v) {
            const int row  = v + 8 * hi;
            const long rof = (long)row * DD + lo;
            float c0 = cbase[rof +  0], s0v = sbase[rof +  0];
            float c1 = cbase[rof + 16], s1v = sbase[rof + 16];
            float c2 = cbase[rof + 32], s2v = sbase[rof + 32];
            float c3 = cbase[rof + 48], s3v = sbase[rof + 48];
            float o0 = a0[v] * c0 - a2[v] * s0v;    // d <  32: rot = -x[d+32]
            float o1 = a1[v] * c1 - a3[v] * s1v;
            float o2 = a2[v] * c2 + a0[v] * s2v;    // d >= 32: rot =  x[d-32]
            float o3 = a3[v] * c3 + a1[v] * s3v;
            ts[row * PAD +  0 + lo] = o0;           // rotated tile back to LDS stage
            ts[row * PAD + 16 + lo] = o1;
            ts[row * PAD + 32 + lo] = o2;
            ts[row * PAD + 48 + lo] = o3;
        }

        // ---- matmul2: d[n] = rot_tile @ Q  (B[k][n] = Qt[16n+lo][k]) ----
        v8f d0 = {}, d1 = {}, d2 = {}, d3 = {};
        #pragma unroll
        for (int kk = 0; kk < 16; ++kk) {
            const int colA = 4 * kk + 2 * hi;
            v2f af = *(const v2f*)(ts + lo * PAD + colA);
            v2f b0 = *(const v2f*)(Qt + (0 * 16 + lo) * PAD + colA);
            v2f b1 = *(const v2f*)(Qt + (1 * 16 + lo) * PAD + colA);
            v2f b2 = *(const v2f*)(Qt + (2 * 16 + lo) * PAD + colA);
            v2f b3 = *(const v2f*)(Qt + (3 * 16 + lo) * PAD + colA);
            d0 = __builtin_amdgcn_wmma_f32_16x16x4_f32(false, af, false, b0, (short)0, d0, false, false);
            d1 = __builtin_amdgcn_wmma_f32_16x16x4_f32(false, af, false, b1, (short)0, d1, false, false);
            d2 = __builtin_amdgcn_wmma_f32_16x16x4_f32(false, af, false, b2, (short)0, d2, false, false);
            d3 = __builtin_amdgcn_wmma_f32_16x16x4_f32(false, af, false, b3, (short)0, d3, false, false);
        }

        // ---- store (C/D layout -> row-major global; 64B-contiguous per half-wave) ----
        #pragma unroll
        for (int v = 0; v < 8; ++v) {
            const int row  = v + 8 * hi;
            const long rof = (long)row * DD + lo;
            gdst[rof +  0] = d0[v];
            gdst[rof + 16] = d1[v];
            gdst[rof + 32] = d2[v];
            gdst[rof + 48] = d3[v];
        }
    }
}

extern "C" void kernel_launch(void* const* d_in, const int* in_sizes, int n_in,
                              void* d_out, int out_size, void* d_ws, size_t ws_size,
                              hipStream_t stream) {
    const float* q    = (const float*)d_in[0];
    const float* k    = (const float*)d_in[1];
    const float* cosp = (const float*)d_in[2];
    const float* sinp = (const float*)d_in[3];
    const float* vs   = (const float*)d_in[4];
    float* out = (float*)d_out;
    float* Qg  = (float*)d_ws;   // 64*64 floats = 16 KB scratch

    RnRoPE_build_Q<<<1, 64, 0, stream>>>(vs, Qg);
    RnRoPE_main<<<1024, 256, 0, stream>>>(q, k, cosp, sinp, Qg, out);
}